// BoxCorrelation_79757542687348
// MI455X (gfx1250) — compile-verified
//
#include <hip/hip_runtime.h>

typedef float v2f __attribute__((ext_vector_type(2)));
typedef float v8f __attribute__((ext_vector_type(8)));

namespace {
constexpr int   kV   = 6;      // views
constexpr int   kP   = 64;     // rois per view
constexpr int   kND  = 32;     // depth bins
constexpr float kW   = 1600.0f;
constexpr float kH   = 928.0f;
constexpr float kEPS = 1e-5f;
constexpr float kDS  = 0.5f;
constexpr float kBS  = (70.0f - 0.5f) / (32.0f * 33.0f);  // LID bin size
}

// One wave32 per point. WMMA f32 16x16x4 computes proj = T(4x4) x pts_cam(4x32)
// for all 6 views x 32 depths of that point.
__global__ __launch_bounds__(256)
void epipolar_box_corr(const float* __restrict__ points,
                       const float* __restrict__ trans,
                       const float* __restrict__ rois,
                       float* __restrict__ out_xy,
                       float* __restrict__ out_mask,
                       float* __restrict__ out_corr,
                       int npts)
{
    // Stage the 6KB ROI table into LDS via the CDNA5 async global->LDS DMA path
    // (GLOBAL_LOAD_ASYNC_TO_LDS_B128, tracked by ASYNCcnt), then fence + barrier.
    __shared__ float4 s_rois[kV * kP];
    {
        const unsigned lds_base = (unsigned)(size_t)(&s_rois[0]);
        for (int i = threadIdx.x; i < kV * kP; i += (int)blockDim.x) {
            const unsigned lds_off = lds_base + (unsigned)i * 16u;
            const unsigned goff    = (unsigned)i * 16u;
            // GVS mode: LDS address from %0, global address is SGPR base %2 plus VGPR offset %1
            asm volatile("global_load_async_to_lds_b128 %0, %1, %2"
                         :: "v"(lds_off), "v"(goff), "s"(rois)
                         : "memory");
        }
        asm volatile("s_wait_asynccnt 0x0" ::: "memory");
    }
    __syncthreads();

    const int n = (int)((blockIdx.x * blockDim.x + threadIdx.x) >> 5);
    if (n >= npts) return;

    const int lane = (int)(threadIdx.x & 31);
    const int j16  = lane & 15;   // column (depth-in-half) / row-in-A
    const int h    = lane >> 4;   // lane half -> K half for A/B, M half for D

    const float px  = points[n * 3 + 1];
    const float py  = points[n * 3 + 2];
    const int   img = (int)points[n * 3 + 0];

    // LID depths for this lane's column (low half: d=j16, high half: d=j16+16)
    const float dep0 = kDS + kBS * (float)j16 * (float)(j16 + 1);
    const float dep1 = kDS + kBS * (float)(j16 + 16) * (float)(j16 + 17);

    // B (4x16): column j = [x*dep, y*dep, dep, 1]^T.
    // Layout: lanes 0-15 hold K=0,1 (vgpr0,1); lanes 16-31 hold K=2,3.
    v2f bLo, bHi;
    if (h == 0) { bLo.x = px * dep0; bLo.y = py * dep0;
                  bHi.x = px * dep1; bHi.y = py * dep1; }
    else        { bLo.x = dep0;      bLo.y = 1.0f;
                  bHi.x = dep1;      bHi.y = 1.0f; }

    // A (16x4): row m = 4*v + i  ->  trans_mats[v][img][i][k].
    // Layout: lane m (m<16) holds K=0,1; lane m+16 holds K=2,3.
    const int vv  = j16 >> 2;   // view within A-set
    const int row = j16 & 3;    // matrix row i
    const int k0  = h << 1;     // K offset 0 or 2
    v2f a0, a1;
    {
        const float* t0 = trans + (size_t)((vv * kV + img) * 16 + row * 4 + k0);
        a0.x = t0[0]; a0.y = t0[1];
        const int vb = vv + 4;                      // views 4,5 (rows 8-15 zero-pad)
        if (vb < kV) {
            const float* t1 = trans + (size_t)((vb * kV + img) * 16 + row * 4 + k0);
            a1.x = t1[0]; a1.y = t1[1];
        } else { a1.x = 0.0f; a1.y = 0.0f; }
    }

    const v8f c0 = {};
    // D[m][j] = dot(T-row m, pts_cam(depth j))
    v8f c00 = __builtin_amdgcn_wmma_f32_16x16x4_f32(false, a0, false, bLo, (short)0, c0, false, false);
    v8f c01 = __builtin_amdgcn_wmma_f32_16x16x4_f32(false, a0, false, bHi, (short)0, c0, false, false);
    v8f c10 = __builtin_amdgcn_wmma_f32_16x16x4_f32(false, a1, false, bLo, (short)0, c0, false, false);
    v8f c11 = __builtin_amdgcn_wmma_f32_16x16x4_f32(false, a1, false, bHi, (short)0, c0, false, false);

    // Killed coords for the box test: invalid -> x = -1e30 so strict '>' fails.
    // Index: qi = a*4 + s*2 + depth_half
    float xq[8], yq[8];

    auto emit = [&](const v8f& c, int a, int dh) {
        const int d = j16 + (dh << 4);
#pragma unroll
        for (int s = 0; s < 2; ++s) {
            // D lane layout: reg r = D[r + 8h][j16]; slot s covers M=4s+8h.. -> view
            const int   v   = (a ? 4 : 0) + (h << 1) + s;
            const float X   = c[4 * s + 0];
            const float Y   = c[4 * s + 1];
            const float Z   = c[4 * s + 2];
            const float inv = 1.0f / fmaxf(Z, kEPS);   // one precise div per slot
            const float xp  = X * inv;
            const float yp  = Y * inv;
            const bool  mk  = (Z > kEPS) & (xp >= 0.0f) & (xp < kW) & (yp >= 0.0f) & (yp < kH);
            const int   qi  = a * 4 + s * 2 + dh;
            if (v < kV) {
                const size_t base = ((size_t)n * kV + v) * kND + d;
                v2f xy; xy.x = xp; xy.y = yp;
                __builtin_nontemporal_store(xy, reinterpret_cast<v2f*>(out_xy) + base);
                __builtin_nontemporal_store(mk ? 1.0f : 0.0f, out_mask + base);
                xq[qi] = mk ? xp : -1.0e30f;
                yq[qi] = yp;
            } else {                 // padded views 6,7 of the A1 set
                xq[qi] = -1.0e30f;
                yq[qi] = 0.0f;
            }
        }
    };
    emit(c00, 0, 0);
    emit(c01, 0, 1);
    emit(c10, 1, 0);
    emit(c11, 1, 1);

    // corr bitmasks per view (wave-uniform values, one bit per box)
    unsigned acc[kV][2];
#pragma unroll
    for (int v = 0; v < kV; ++v) { acc[v][0] = 0u; acc[v][1] = 0u; }

    for (int p = 0; p < kP; ++p) {
        const unsigned bit = 1u << (p & 31);
        const int ph = p >> 5;
#pragma unroll
        for (int a = 0; a < 2; ++a) {
#pragma unroll
            for (int s = 0; s < 2; ++s) {
                const int v  = (a ? 4 : 0) + (h << 1) + s;
                const int vc = (v < kV) ? v : 0;
                const float4 bx = s_rois[vc * kP + p];   // [x1,y1,x2,y2]
                const int q = a * 4 + s * 2;
                const bool pr =
                    ((xq[q]     > bx.x) & (xq[q]     < bx.z) & (yq[q]     > bx.y) & (yq[q]     < bx.w)) |
                    ((xq[q + 1] > bx.x) & (xq[q + 1] < bx.z) & (yq[q + 1] > bx.y) & (yq[q + 1] < bx.w));
                // bits 0-15: view (base+s), bits 16-31: view (base+s+2)
                const unsigned bal = __builtin_amdgcn_ballot_w32(pr);
                const int vlo = (a ? 4 : 0) + s;
                const int vhi = vlo + 2;     // a=1 -> 6,7: compile-time dead
                if (bal & 0xFFFFu) acc[vlo][ph] |= bit;
                if (vhi < kV) { if (bal >> 16) acc[vhi][ph] |= bit; }
            }
        }
    }

    // Write corr as 12 fully coalesced 32-lane stores.
#pragma unroll
    for (int i = 0; i < 12; ++i) {
        const int v  = i >> 1;
        const int ph = i & 1;
        const int p  = (ph << 5) | lane;
        const float val = ((acc[v][ph] >> lane) & 1u) ? 1.0f : 0.0f;
        __builtin_nontemporal_store(val, out_corr + ((size_t)n * kV + v) * kP + p);
    }
}

extern "C" void kernel_launch(void* const* d_in, const int* in_sizes, int n_in,
                              void* d_out, int out_size, void* d_ws, size_t ws_size,
                              hipStream_t stream) {
    const float* points = (const float*)d_in[0];
    const float* trans  = (const float*)d_in[1];
    const float* rois   = (const float*)d_in[2];
    const int npts = in_sizes[0] / 3;

    float* out_xy   = (float*)d_out;                          // (N,V,32,2)
    float* out_mask = out_xy   + (size_t)npts * kV * kND * 2; // (N,V,32)
    float* out_corr = out_mask + (size_t)npts * kV * kND;     // (N,V,64)

    const int waves_per_block = 256 / 32;
    const int blocks = (npts + waves_per_block - 1) / waves_per_block;
    hipLaunchKernelGGL(epipolar_box_corr, dim3(blocks), dim3(256), 0, stream,
                       points, trans, rois, out_xy, out_mask, out_corr, npts);
}